// RegionProposalNetwork_56624848830840
// MI455X (gfx1250) — compile-verified
//
#include <hip/hip_runtime.h>
#include <math.h>

typedef __attribute__((ext_vector_type(2))) float v2f;
typedef __attribute__((ext_vector_type(8))) float v8f;

#define BB 8
#define AAN 3
#define NTOT 159882
#define KTOT 4507
#define KPAD 8192
#define JW 141            // ceil(KTOT/32)
#define POST 1000
#define T16 282           // ceil(KTOT/16)
#define PAIRS 39903       // T16*(T16+1)/2
#define IMGSZ 800.0f
#define MIN_SZ 1e-3f
#define NMS_TH_C 0.7f
#define LVL_OFF_F 4096.0f
#define BBOX_CLIP 4.135166556742356f

__device__ __forceinline__ int lvl_of(int n) {
  if (n < 120000) return 0;
  if (n < 150000) return 1;
  if (n < 157500) return 2;
  if (n < 159375) return 3;
  return 4;
}
__device__ __forceinline__ int lvl_off(int l) {
  switch (l) { case 0: return 0; case 1: return 120000; case 2: return 150000;
               case 3: return 157500; default: return 159375; }
}
__device__ __forceinline__ int lvl_h(int l) {
  switch (l) { case 0: return 200; case 1: return 100; case 2: return 50;
               case 3: return 25; default: return 13; }
}
__device__ __forceinline__ int lvl_n(int l) {
  switch (l) { case 0: return 120000; case 1: return 30000; case 2: return 7500;
               case 3: return 1875; default: return 507; }
}
__device__ __forceinline__ int lvl_k(int l) { return (l == 4) ? 507 : 1000; }
__device__ __forceinline__ int lvl_koff(int l) { return l * 1000; }

// monotonic float->uint key (descending top-k == largest keys)
__device__ __forceinline__ unsigned fkey(float f) {
  unsigned u = __float_as_uint(f);
  return (u & 0x80000000u) ? ~u : (u | 0x80000000u);
}

// ---------------- 1. flatten objectness to [B, N] in (y,x,a) order ----------
__global__ void k_flatten(const float* __restrict__ o0, const float* __restrict__ o1,
                          const float* __restrict__ o2, const float* __restrict__ o3,
                          const float* __restrict__ o4, float* __restrict__ scores) {
  int g = blockIdx.x * blockDim.x + threadIdx.x;
  if (g >= BB * NTOT) return;
  int b = g / NTOT, n = g % NTOT;
  int l = lvl_of(n);
  int m = n - lvl_off(l);
  int a = m % AAN, pix = m / AAN;
  int h = lvl_h(l);
  int x = pix % h, y = pix / h;
  const float* o = (l == 0) ? o0 : (l == 1) ? o1 : (l == 2) ? o2 : (l == 3) ? o3 : o4;
  scores[g] = o[(((size_t)b * AAN + a) * h + y) * h + x];
}

// ---------------- 2. per-(b,lvl) radix select of k-th largest ---------------
__global__ void k_radix(const float* __restrict__ scores, unsigned* __restrict__ Tout,
                        unsigned* __restrict__ CG, unsigned* __restrict__ cntH,
                        unsigned* __restrict__ cntE) {
  int bl = blockIdx.x;           // b*5 + lvl
  int b = bl / 5, lvl = bl % 5;
  const float* sc = scores + (size_t)b * NTOT + lvl_off(lvl);
  int n = lvl_n(lvl), k = lvl_k(lvl);
  __shared__ unsigned hist[256];
  __shared__ unsigned s_prefix, s_cntG;
  __shared__ int s_remK;
  if (threadIdx.x == 0) { s_prefix = 0u; s_cntG = 0u; s_remK = k; }
  __syncthreads();
  for (int byte = 3; byte >= 0; --byte) {
    hist[threadIdx.x] = 0u;
    __syncthreads();
    unsigned pfx = s_prefix;
    int hishift = (byte + 1) * 8;
    for (int i = threadIdx.x; i < n; i += 256) {
      unsigned key = fkey(sc[i]);
      bool cond = (byte == 3) || ((key >> hishift) == (pfx >> hishift));
      if (cond) atomicAdd(&hist[(key >> (byte * 8)) & 255u], 1u);
    }
    __syncthreads();
    if (threadIdx.x == 0) {
      int rem = s_remK;
      unsigned cum = 0;
      int chosen = 0;
      for (int v = 255; v >= 0; --v) {
        unsigned hc = hist[v];
        if (cum + hc >= (unsigned)rem) { chosen = v; break; }
        cum += hc;
      }
      s_cntG += cum;
      s_remK = rem - (int)cum;
      s_prefix = pfx | ((unsigned)chosen << (byte * 8));
    }
    __syncthreads();
  }
  if (threadIdx.x == 0) {
    Tout[bl] = s_prefix;
    CG[bl] = s_cntG;
    cntH[bl] = 0u;
    cntE[bl] = 0u;
  }
}

// ---------------- 3. compact: write top-k indices per (b,lvl) ---------------
__global__ void k_compact(const float* __restrict__ scores, const unsigned* __restrict__ Tin,
                          const unsigned* __restrict__ CG, unsigned* __restrict__ cntH,
                          unsigned* __restrict__ cntE, int* __restrict__ cand_idx) {
  int g = blockIdx.x * blockDim.x + threadIdx.x;
  if (g >= BB * NTOT) return;
  int b = g / NTOT, n = g % NTOT;
  int lvl = lvl_of(n);
  int bl = b * 5 + lvl;
  unsigned key = fkey(scores[g]);
  unsigned Tv = Tin[bl];
  if (key > Tv) {
    unsigned p = atomicAdd(&cntH[bl], 1u);
    cand_idx[(size_t)b * KTOT + lvl_koff(lvl) + p] = n;
  } else if (key == Tv) {
    unsigned p = atomicAdd(&cntE[bl], 1u);
    unsigned s = CG[bl] + p;
    if (s < (unsigned)lvl_k(lvl))
      cand_idx[(size_t)b * KTOT + lvl_koff(lvl) + s] = n;
  }
}

// ---------------- 4. decode + clip + valid for K candidates ----------------
__global__ void k_decode(const int* __restrict__ cand_idx, const float* __restrict__ scores,
                         const float* __restrict__ anchors,
                         const float* __restrict__ d0, const float* __restrict__ d1,
                         const float* __restrict__ d2, const float* __restrict__ d3,
                         const float* __restrict__ d4,
                         float* __restrict__ ckey, float* __restrict__ cout,
                         float* __restrict__ cshift) {
  int g = blockIdx.x * blockDim.x + threadIdx.x;
  if (g >= BB * KTOT) return;
  int b = g / KTOT;
  int idx = cand_idx[g];
  int l = lvl_of(idx);
  int m = idx - lvl_off(l);
  int a = m % AAN, pix = m / AAN;
  int h = lvl_h(l);
  int x = pix % h, y = pix / h;
  size_t hh = (size_t)h * h;
  const float* dp = (l == 0) ? d0 : (l == 1) ? d1 : (l == 2) ? d2 : (l == 3) ? d3 : d4;
  size_t base = ((size_t)(b * AAN + a) * 4) * hh + (size_t)y * h + x;
  float dx = dp[base], dy = dp[base + hh];
  float dw = fminf(dp[base + 2 * hh], BBOX_CLIP);
  float dh = fminf(dp[base + 3 * hh], BBOX_CLIP);
  const float* an = anchors + (size_t)idx * 4;
  float wa = an[2] - an[0], ha = an[3] - an[1];
  float cxa = an[0] + 0.5f * wa, cya = an[1] + 0.5f * ha;
  float cx = dx * wa + cxa, cy = dy * ha + cya;
  float pw = __expf(dw) * wa, ph = __expf(dh) * ha;
  float x1 = fminf(fmaxf(cx - 0.5f * pw, 0.0f), IMGSZ);
  float y1 = fminf(fmaxf(cy - 0.5f * ph, 0.0f), IMGSZ);
  float x2 = fminf(fmaxf(cx + 0.5f * pw, 0.0f), IMGSZ);
  float y2 = fminf(fmaxf(cy + 0.5f * ph, 0.0f), IMGSZ);
  bool valid = ((x2 - x1) >= MIN_SZ) && ((y2 - y1) >= MIN_SZ);
  float logit = scores[(size_t)b * NTOT + idx];
  float sc = 1.0f / (1.0f + __expf(-logit));
  ckey[g] = valid ? sc : -INFINITY;
  float* ob = cout + (size_t)g * 4;
  ob[0] = x1; ob[1] = y1; ob[2] = x2; ob[3] = y2;
  float off = (float)l * LVL_OFF_F;
  float* sb = cshift + (size_t)g * 4;
  sb[0] = x1 + off; sb[1] = y1 + off; sb[2] = x2 + off; sb[3] = y2 + off;
}

// ---------------- 5. per-image bitonic sort (desc by key) in LDS ------------
__global__ void __launch_bounds__(1024)
k_sort(const float* __restrict__ ckey, const float* __restrict__ cout,
       const float* __restrict__ cshift, float* __restrict__ sout,
       float* __restrict__ sshift, int* __restrict__ svalid) {
  extern __shared__ char smem[];
  float* skey = (float*)smem;
  int* sidx = (int*)(smem + KPAD * sizeof(float));
  int b = blockIdx.x;
  int tid = threadIdx.x;
  for (int t = tid; t < KPAD; t += 1024) {
    if (t < KTOT) { skey[t] = ckey[(size_t)b * KTOT + t]; sidx[t] = t; }
    else          { skey[t] = -INFINITY;                   sidx[t] = -1; }
  }
  __syncthreads();
  for (unsigned size = 2; size <= KPAD; size <<= 1) {
    for (unsigned stride = size >> 1; stride > 0; stride >>= 1) {
      for (unsigned t = tid; t < KPAD / 2; t += 1024) {
        unsigned i = ((t / stride) * stride * 2u) + (t % stride);
        unsigned j = i + stride;
        bool desc = ((i & size) == 0u);
        float ki = skey[i], kj = skey[j];
        bool sw = desc ? (ki < kj) : (ki > kj);
        if (sw) {
          skey[i] = kj; skey[j] = ki;
          int tt = sidx[i]; sidx[i] = sidx[j]; sidx[j] = tt;
        }
      }
      __syncthreads();
    }
  }
  for (int t = tid; t < KTOT; t += 1024) {
    int idx = sidx[t];
    float kk = skey[t];
    bool val = (idx >= 0) && (kk > -1e30f);
    float* od = sout + ((size_t)b * KTOT + t) * 4;
    float* sd = sshift + ((size_t)b * KTOT + t) * 4;
    if (val) {
      const float* os = cout + ((size_t)b * KTOT + idx) * 4;
      const float* ss = cshift + ((size_t)b * KTOT + idx) * 4;
      od[0] = os[0]; od[1] = os[1]; od[2] = os[2]; od[3] = os[3];
      sd[0] = ss[0]; sd[1] = ss[1]; sd[2] = ss[2]; sd[3] = ss[3];
    } else {
      od[0] = od[1] = od[2] = od[3] = 0.0f;
      sd[0] = sd[1] = sd[2] = sd[3] = 0.0f;
    }
    svalid[(size_t)b * KTOT + t] = val ? 1 : 0;
  }
}

// ---------------- 6. K x K IoU suppression bit-matrix, 16x16 WMMA tiles -----
// One wave32 per 16x16 tile. Union term area_i + area_j computed for the whole
// tile with a single V_WMMA_F32_16X16X4_F32 rank-2 outer product; row boxes
// staged in LDS; division-free threshold test; suppression bits combined
// per-row with wave ballots (2 rows per ballot, matching the WMMA C/D lane
// layout), 16 atomics per tile.
__global__ void k_iou(const float* __restrict__ sshift, unsigned* __restrict__ mask) {
  int p = blockIdx.x % PAIRS;
  int b = blockIdx.x / PAIRS;
  // decode triangular pair (ti, tj), tj >= ti
  int ti = 0, rowoff = 0;
  while (p >= rowoff + (T16 - ti)) { rowoff += (T16 - ti); ti++; }
  int tj = ti + (p - rowoff);
  int ri = ti * 16, cj = tj * 16;
  const float* Sb = sshift + (size_t)b * KTOT * 4;
  int lane = threadIdx.x;
  int lm = lane & 15;
  __shared__ float srow[16][4];
  float areaR = 0.0f, areaC = 0.0f;
  if (lane < 16) {
    int r = ri + lm;
    bool rok = (r < KTOT);
    const float* bxr = Sb + (size_t)(rok ? r : (KTOT - 1)) * 4;
    float b0 = bxr[0], b1 = bxr[1], b2 = bxr[2], b3 = bxr[3];
    if (!rok) { b0 = b1 = b2 = b3 = 0.0f; }
    srow[lm][0] = b0; srow[lm][1] = b1; srow[lm][2] = b2; srow[lm][3] = b3;
    areaR = (b2 - b0) * (b3 - b1);
    int c = cj + lm;
    bool cok = (c < KTOT);
    const float* bxc = Sb + (size_t)(cok ? c : (KTOT - 1)) * 4;
    float c0 = bxc[0], c1 = bxc[1], c2 = bxc[2], c3 = bxc[3];
    areaC = cok ? (c2 - c0) * (c3 - c1) : 0.0f;
  }
  __syncthreads();
  // outer-product union precompute: D[i][j] = area_i*1 + 1*area_j
  // A (16x4 f32): col0 = area_i, col1 = 1, cols 2..3 = 0 (lanes 16-31 zero)
  // B (4x16 f32): row0 = 1,      row1 = area_j, rows 2..3 = 0
  v2f av, bv;
  av.x = (lane < 16) ? areaR : 0.0f;
  av.y = (lane < 16) ? 1.0f : 0.0f;
  bv.x = (lane < 16) ? 1.0f : 0.0f;
  bv.y = (lane < 16) ? areaC : 0.0f;
  v8f acc = {0.f, 0.f, 0.f, 0.f, 0.f, 0.f, 0.f, 0.f};
  acc = __builtin_amdgcn_wmma_f32_16x16x4_f32(false, av, false, bv,
                                              (short)0, acc, false, false);
  int j = cj + lm;
  bool jok = (j < KTOT);
  const float* bxj = Sb + (size_t)(jok ? j : (KTOT - 1)) * 4;
  float jx1 = bxj[0], jy1 = bxj[1], jx2 = bxj[2], jy2 = bxj[3];
  int mbase = (lane >= 16) ? 8 : 0;
  unsigned halfshift = (unsigned)(cj & 16);
  size_t jw = (size_t)(cj >> 5);
  unsigned* mrow_base = mask + (size_t)b * KTOT * JW + jw;
#pragma unroll
  for (int v = 0; v < 8; ++v) {
    int mrow = mbase + v;
    int i = ri + mrow;
    float bi0 = srow[mrow][0], bi1 = srow[mrow][1];
    float bi2 = srow[mrow][2], bi3 = srow[mrow][3];
    float lx = fmaxf(bi0, jx1), ly = fmaxf(bi1, jy1);
    float rx = fminf(bi2, jx2), ry = fminf(bi3, jy2);
    float w = fmaxf(rx - lx, 0.0f), hgt = fmaxf(ry - ly, 0.0f);
    float inter = w * hgt;
    float uni = acc[v] - inter;         // area_i + area_j - inter (from WMMA)
    // iou > th  <=>  inter > th*(union+eps)   (union+eps > 0 always)
    bool pred = jok && (j > i) && (i < KTOT) &&
                (inter > NMS_TH_C * (uni + 1e-9f));
    unsigned bal = (unsigned)__ballot(pred);
    // bits [15:0] -> row ri+v (lanes 0-15), bits [31:16] -> row ri+v+8
    if (lane == 0) {
      unsigned bits = bal & 0xFFFFu;
      if (bits && (ri + v) < KTOT)
        atomicOr(mrow_base + (size_t)(ri + v) * JW, bits << halfshift);
    } else if (lane == 16) {
      unsigned bits = (bal >> 16) & 0xFFFFu;
      if (bits && (ri + v + 8) < KTOT)
        atomicOr(mrow_base + (size_t)(ri + v + 8) * JW, bits << halfshift);
    }
  }
}

// ---------------- 7. serial greedy scan per image + emit output -------------
__global__ void k_scan(const float* __restrict__ sout, const int* __restrict__ svalid,
                       const unsigned* __restrict__ mask, float* __restrict__ out) {
  int b = blockIdx.x;
  int tid = threadIdx.x;
  __shared__ unsigned sup[JW];
  __shared__ int s_keep, s_cnt;
  for (int w = tid; w < JW; w += blockDim.x) {
    unsigned mword = 0;
    for (int bit = 0; bit < 32; ++bit) {
      int i = w * 32 + bit;
      if (i < KTOT && svalid[(size_t)b * KTOT + i] == 0) mword |= (1u << bit);
    }
    sup[w] = mword;
  }
  if (tid == 0) s_cnt = 0;
  __syncthreads();
  for (int i = 0; i < KTOT; ++i) {
    if (tid == 0) {
      int kp = ((sup[i >> 5] >> (i & 31)) & 1u) == 0u;
      s_keep = kp;
      if (kp) {
        if (s_cnt < POST) {
          const float* src = sout + ((size_t)b * KTOT + i) * 4;
          float* dst = out + ((size_t)b * POST + s_cnt) * 4;
          dst[0] = src[0]; dst[1] = src[1]; dst[2] = src[2]; dst[3] = src[3];
        }
        s_cnt++;
      }
    }
    __syncthreads();
    if (s_keep) {
      const unsigned* mr = mask + ((size_t)b * KTOT + i) * JW;
      for (int w = tid; w < JW; w += blockDim.x) sup[w] |= mr[w];
    }
    __syncthreads();
  }
  int c = s_cnt;
  if (c > POST) c = POST;
  for (int t = tid; t < (POST - c) * 4; t += blockDim.x)
    out[((size_t)b * POST + c) * 4 + t] = 0.0f;
}

extern "C" void kernel_launch(void* const* d_in, const int* in_sizes, int n_in,
                              void* d_out, int out_size, void* d_ws, size_t ws_size,
                              hipStream_t stream) {
  // setup_inputs() dict order: obj0, delta0, obj1, delta1, ..., obj4, delta4, anchors
  const float* o0 = (const float*)d_in[0];
  const float* de0 = (const float*)d_in[1];
  const float* o1 = (const float*)d_in[2];
  const float* de1 = (const float*)d_in[3];
  const float* o2 = (const float*)d_in[4];
  const float* de2 = (const float*)d_in[5];
  const float* o3 = (const float*)d_in[6];
  const float* de3 = (const float*)d_in[7];
  const float* o4 = (const float*)d_in[8];
  const float* de4 = (const float*)d_in[9];
  const float* anchors = (const float*)d_in[10];

  char* p = (char*)d_ws;
  auto alloc = [&](size_t bytes) -> void* {
    void* r = (void*)p;
    p += (bytes + 255) & ~(size_t)255;
    return r;
  };
  float* scores     = (float*)alloc((size_t)BB * NTOT * sizeof(float));
  unsigned* Tarr    = (unsigned*)alloc(40 * sizeof(unsigned));
  unsigned* CG      = (unsigned*)alloc(40 * sizeof(unsigned));
  unsigned* cntH    = (unsigned*)alloc(40 * sizeof(unsigned));
  unsigned* cntE    = (unsigned*)alloc(40 * sizeof(unsigned));
  int* cand_idx     = (int*)alloc((size_t)BB * KTOT * sizeof(int));
  float* ckey       = (float*)alloc((size_t)BB * KTOT * sizeof(float));
  float* cout_      = (float*)alloc((size_t)BB * KTOT * 4 * sizeof(float));
  float* cshift     = (float*)alloc((size_t)BB * KTOT * 4 * sizeof(float));
  float* sout       = (float*)alloc((size_t)BB * KTOT * 4 * sizeof(float));
  float* sshift     = (float*)alloc((size_t)BB * KTOT * 4 * sizeof(float));
  int* svalid       = (int*)alloc((size_t)BB * KTOT * sizeof(int));
  size_t maskBytes  = (size_t)BB * KTOT * JW * sizeof(unsigned);
  unsigned* mask    = (unsigned*)alloc(maskBytes);

  int totBN = BB * NTOT;
  k_flatten<<<(totBN + 255) / 256, 256, 0, stream>>>(o0, o1, o2, o3, o4, scores);
  k_radix<<<40, 256, 0, stream>>>(scores, Tarr, CG, cntH, cntE);
  k_compact<<<(totBN + 255) / 256, 256, 0, stream>>>(scores, Tarr, CG, cntH, cntE, cand_idx);
  k_decode<<<(BB * KTOT + 255) / 256, 256, 0, stream>>>(cand_idx, scores, anchors,
                                                        de0, de1, de2, de3, de4,
                                                        ckey, cout_, cshift);
  k_sort<<<BB, 1024, KPAD * (sizeof(float) + sizeof(int)), stream>>>(
      ckey, cout_, cshift, sout, sshift, svalid);
  hipMemsetAsync(mask, 0, maskBytes, stream);
  k_iou<<<BB * PAIRS, 32, 0, stream>>>(sshift, mask);
  k_scan<<<BB, 256, 0, stream>>>(sout, svalid, mask, (float*)d_out);
}